// PseudoMaskLoss_75565654605927
// MI455X (gfx1250) — compile-verified
//
#include <hip/hip_runtime.h>

// Problem constants (from reference): K slots, N images, L pixels, 16 labels.
#define KSLOTS 8
#define NIMG   64
#define LPIX   65536
#define NLAB   16
#define NSEG   (NIMG * NLAB)          // 1024 segments
#define NWAVES 8                      // 256 threads / wave32

typedef __attribute__((ext_vector_type(2))) float v2f;
typedef __attribute__((ext_vector_type(8))) float v8f;

// ---------------------------------------------------------------------------
// Zero the 8 KB of scratch (sq[1024] f32 + cnt[1024] u32) each call.
// ---------------------------------------------------------------------------
__global__ void pml_zero_ws(unsigned* __restrict__ ws) {
    int i = blockIdx.x * blockDim.x + threadIdx.x;
    if (i < 2 * NSEG) ws[i] = 0u;
}

// ---------------------------------------------------------------------------
// Main bandwidth pass: each block owns 1024 contiguous pixels of one image.
// Per thread: 1x int4 mask load + 8x float4 soft loads (all global_load_b128,
// fully coalesced per k-plane), col = sum over k, then LDS per-wave binning.
// ---------------------------------------------------------------------------
__global__ __launch_bounds__(256)
void pml_accum(const int* __restrict__ pmask,
               const float* __restrict__ soft,
               float* __restrict__ g_sq,
               unsigned* __restrict__ g_cnt) {
    __shared__ float    s_sq[NWAVES][NLAB];
    __shared__ unsigned s_cnt[NWAVES][NLAB];

    const int tid = threadIdx.x;
    if (tid < NWAVES * NLAB) {
        (&s_sq[0][0])[tid]  = 0.0f;
        (&s_cnt[0][0])[tid] = 0u;
    }
    __syncthreads();

    // 64 blocks per image (65536 / 1024), 4096 blocks total.
    const int    n    = blockIdx.x >> 6;
    const int    l    = ((blockIdx.x & 63) << 10) + (tid << 2);
    const size_t base = (size_t)n * LPIX + (size_t)l;

    const int4 m4 = *(const int4*)(pmask + base);

    float4 col = make_float4(0.f, 0.f, 0.f, 0.f);
#pragma unroll
    for (int k = 0; k < KSLOTS; ++k) {
        const float4 v = *(const float4*)(soft + (size_t)k * ((size_t)NIMG * LPIX) + base);
        col.x += v.x; col.y += v.y; col.z += v.z; col.w += v.w;
    }

    const int w = tid >> 5;   // wave id within block
    atomicAdd(&s_cnt[w][m4.x & (NLAB - 1)], 1u);
    atomicAdd(&s_sq [w][m4.x & (NLAB - 1)], col.x * col.x);
    atomicAdd(&s_cnt[w][m4.y & (NLAB - 1)], 1u);
    atomicAdd(&s_sq [w][m4.y & (NLAB - 1)], col.y * col.y);
    atomicAdd(&s_cnt[w][m4.z & (NLAB - 1)], 1u);
    atomicAdd(&s_sq [w][m4.z & (NLAB - 1)], col.z * col.z);
    atomicAdd(&s_cnt[w][m4.w & (NLAB - 1)], 1u);
    atomicAdd(&s_sq [w][m4.w & (NLAB - 1)], col.w * col.w);
    __syncthreads();

    // One thread per label: fold the 8 wave-private bins, push to global.
    if (tid < NLAB) {
        float    sq = 0.0f;
        unsigned c  = 0u;
#pragma unroll
        for (int wv = 0; wv < NWAVES; ++wv) {
            sq += s_sq[wv][tid];
            c  += s_cnt[wv][tid];
        }
        atomicAdd(&g_sq [n * NLAB + tid], sq);
        atomicAdd(&g_cnt[n * NLAB + tid], c);
    }
}

// ---------------------------------------------------------------------------
// Finalize over 1024 segments with a single wave32. The 32 per-lane partials
// are collapsed with V_WMMA_F32_16X16X4_F32 against a ones B-matrix:
//   A (16x4 f32, striped: lanes 0-15 hold K=0,1; lanes 16-31 hold K=2,3)
//   carries partial in K=0/2 and 0 in K=1/3, B = ones -> D[m,*] =
//   partial[m] + partial[m+16]. Summing the 8 D VGPRs gives the M=0..7 total
//   in lanes 0-15 and the M=8..15 total in lanes 16-31; one shuffle combine
//   finishes the reduction. EXEC is all-1s at the WMMA (single full wave).
// ---------------------------------------------------------------------------
__global__ __launch_bounds__(32)
void pml_finalize(const float* __restrict__ g_sq,
                  const unsigned* __restrict__ g_cnt,
                  float* __restrict__ out) {
    const int lane = threadIdx.x;

    float partial = 0.0f;
    for (int s = lane; s < NSEG; s += 32) {
        const unsigned c  = g_cnt[s];
        const float    sq = g_sq[s];
        const bool valid  = (c > 1u) && ((s & (NLAB - 1)) != 0);
        partial += valid ? ((float)(KSLOTS * KSLOTS) - sq) : 0.0f;
    }

    v2f a; a[0] = partial; a[1] = 0.0f;   // A: partial at K=0 (or K=2), 0 at K=1 (or K=3)
    v2f b; b[0] = 1.0f;    b[1] = 1.0f;   // B: all ones
    v8f acc = {0.f, 0.f, 0.f, 0.f, 0.f, 0.f, 0.f, 0.f};
    acc = __builtin_amdgcn_wmma_f32_16x16x4_f32(
        /*neg_a=*/false, a, /*neg_b=*/false, b,
        /*c_mod=*/(short)0, acc, /*reuse_a=*/false, /*reuse_b=*/false);

    float row = acc[0] + acc[1] + acc[2] + acc[3] +
                acc[4] + acc[5] + acc[6] + acc[7];

    const float lo = __shfl(row, 0, 32);   // M = 0..7 total (lanes 0-15)
    const float hi = __shfl(row, 16, 32);  // M = 8..15 total (lanes 16-31)

    if (lane == 0) {
        // 1 / (N * L * L) = 2^-38, exact in f32.
        out[0] = (lo + hi) * 3.637978807091713e-12f;
    }
}

// ---------------------------------------------------------------------------
extern "C" void kernel_launch(void* const* d_in, const int* in_sizes, int n_in,
                              void* d_out, int out_size, void* d_ws, size_t ws_size,
                              hipStream_t stream) {
    (void)in_sizes; (void)n_in; (void)out_size; (void)ws_size;

    const int*   pmask = (const int*)d_in[0];    // (N, L) int32
    const float* soft  = (const float*)d_in[1];  // (K, N, L) float32

    float*    g_sq  = (float*)d_ws;
    unsigned* g_cnt = (unsigned*)((char*)d_ws + NSEG * sizeof(float));

    pml_zero_ws<<<(2 * NSEG + 255) / 256, 256, 0, stream>>>((unsigned*)d_ws);

    const int blocks = NIMG * (LPIX / 1024);     // 4096 blocks, 1024 px each
    pml_accum<<<blocks, 256, 0, stream>>>(pmask, soft, g_sq, g_cnt);

    pml_finalize<<<1, 32, 0, stream>>>(g_sq, g_cnt, (float*)d_out);
}